// GNNEncoder_4715874091025
// MI455X (gfx1250) — compile-verified
//
#include <hip/hip_runtime.h>

#define N_NODES 50000
#define N_EDGES 800000
#define F_IN    32
#define H       128
#define LN_EPS  1e-5f

#define ROWS    80          // nodes per block in gcn_layer (50000 = 625 * 80)
#define NSUB    (ROWS / 16) // 5 row sub-tiles of 16, sharing one B fragment
#define HP      (H + 4)     // padded row stride: 528B keeps 16B alignment, bank stride 4

typedef float v2f __attribute__((ext_vector_type(2)));
typedef float v8f __attribute__((ext_vector_type(8)));

// ---------------- zero scratch ----------------
__global__ void zero_f32(float* __restrict__ p, int n) {
    int i = blockIdx.x * blockDim.x + threadIdx.x;
    if (i < n) p[i] = 0.0f;
}

// ---------------- in-degree counts (fp32 so divide is cheap later) ----------------
__global__ void count_edges(const int* __restrict__ tgt, float* __restrict__ cnt) {
    int e = blockIdx.x * blockDim.x + threadIdx.x;
    if (e < N_EDGES)
        __hip_atomic_fetch_add(&cnt[tgt[e]], 1.0f,
                               __ATOMIC_RELAXED, __HIP_MEMORY_SCOPE_AGENT);
}

// ---------------- edge gather + scatter-add: one wave32 per edge ----------------
// 128 floats/edge -> float4 per lane; agg (25.6MB) is L2-resident (192MB L2) so
// the global_atomic_add_f32 traffic stays on-chip.
__global__ __launch_bounds__(256) void edge_aggregate(const float* __restrict__ h,
                                                      const int* __restrict__ src,
                                                      const int* __restrict__ tgt,
                                                      float* __restrict__ agg) {
    int wid  = (blockIdx.x * blockDim.x + threadIdx.x) >> 5;   // edge id
    int lane = threadIdx.x & 31;
    if (wid >= N_EDGES) return;
    int s = src[wid];
    int d = tgt[wid];
    const float4 v = *(const float4*)(h + (size_t)s * H + lane * 4);
    float* p = agg + (size_t)d * H + lane * 4;
    __hip_atomic_fetch_add(p + 0, v.x, __ATOMIC_RELAXED, __HIP_MEMORY_SCOPE_AGENT);
    __hip_atomic_fetch_add(p + 1, v.y, __ATOMIC_RELAXED, __HIP_MEMORY_SCOPE_AGENT);
    __hip_atomic_fetch_add(p + 2, v.z, __ATOMIC_RELAXED, __HIP_MEMORY_SCOPE_AGENT);
    __hip_atomic_fetch_add(p + 3, v.w, __ATOMIC_RELAXED, __HIP_MEMORY_SCOPE_AGENT);
}

// ---------------- input projection + ReLU via V_WMMA_F32_16X16X4_F32 ----------------
// Block = 16 nodes x 128 outputs; 8 waves, each wave owns a 16x16 output tile.
__global__ __launch_bounds__(256) void proj_relu(const float* __restrict__ x,   // [N][F_IN]
                                                 const float* __restrict__ w,   // [F_IN][H]
                                                 const float* __restrict__ b,   // [H]
                                                 float* __restrict__ out) {     // [N][H]
    __shared__ float Ash[16][F_IN + 4];
    int t = threadIdx.x;
    int node0 = blockIdx.x * 16;
    for (int i = t; i < 16 * F_IN; i += 256) {
        int r = i >> 5, c = i & (F_IN - 1);
        Ash[r][c] = x[(size_t)(node0 + r) * F_IN + c];
    }
    __syncthreads();

    int warp = t >> 5, lane = t & 31;
    int hh = lane >> 4, lr = lane & 15;       // half-wave select, row/col in tile
    int col = warp * 16 + lr;

    v8f acc = {};
    for (int kc = 0; kc < F_IN / 4; ++kc) {
        int k = kc * 4 + 2 * hh;              // lanes 0-15: K,K+1 ; lanes 16-31: K+2,K+3
        v2f a, bb;
        a.x  = Ash[lr][k];        a.y  = Ash[lr][k + 1];
        bb.x = w[k * H + col];    bb.y = w[(k + 1) * H + col];
        acc = __builtin_amdgcn_wmma_f32_16x16x4_f32(false, a, false, bb,
                                                    (short)0, acc, false, false);
    }
    float bias = b[col];
    #pragma unroll
    for (int v = 0; v < 8; ++v) {
        int row = v + 8 * hh;                 // D layout: VGPR v -> row v + 8*half
        out[(size_t)(node0 + row) * H + col] = fmaxf(acc[v] + bias, 0.0f);
    }
}

// ---------------- fused GCN layer: dual WMMA GEMM + bias + LayerNorm (+ReLU+res) ----
// 80-node tile: each wave reuses one B fragment across 5 wmma issues per chain,
// cutting per-layer weight re-read traffic from ~400MB to ~80MB of L2.
__global__ __launch_bounds__(256) void gcn_layer(const float* __restrict__ hin,
                                                 const float* __restrict__ agg,
                                                 const float* __restrict__ cnt,
                                                 const float* __restrict__ Wself,
                                                 const float* __restrict__ bself,
                                                 const float* __restrict__ Wneigh,
                                                 const float* __restrict__ bneigh,
                                                 const float* __restrict__ gamma,
                                                 const float* __restrict__ beta,
                                                 float* __restrict__ out,
                                                 int relu_residual) {
    __shared__ float Ash[ROWS][HP];   // h tile (self path + residual source)
    __shared__ float Nsh[ROWS][HP];   // mean-aggregated neighbor tile
    __shared__ float Osh[ROWS][HP];   // pre-LN output tile
    __shared__ float bsh[H];          // combined bias bs+bn
    int t = threadIdx.x;
    int node0 = blockIdx.x * ROWS;

    if (t < H) bsh[t] = bself[t] + bneigh[t];
    // float4 cooperative fill: ROWS*32 float4s, 10 per thread
    for (int i = t; i < ROWS * (H / 4); i += 256) {
        int r = i >> 5, c4 = (i & 31) * 4;
        size_t idx = (size_t)(node0 + r) * H + c4;
        float4 hv = *(const float4*)(hin + idx);
        float4 av = *(const float4*)(agg + idx);
        float inv = 1.0f / fmaxf(cnt[node0 + r], 1.0f);   // uniform per row, L2-hot
        *(float4*)&Ash[r][c4] = hv;
        float4 nv = {av.x * inv, av.y * inv, av.z * inv, av.w * inv};
        *(float4*)&Nsh[r][c4] = nv;
    }
    __syncthreads();

    int warp = t >> 5, lane = t & 31;
    int hh = lane >> 4, lr = lane & 15;
    int col = warp * 16 + lr;

    v8f cs[NSUB], cn[NSUB];
    #pragma unroll
    for (int s = 0; s < NSUB; ++s) { cs[s] = (v8f){}; cn[s] = (v8f){}; }

    for (int kc = 0; kc < H / 4; ++kc) {      // 32 K-steps, 10 wmma each
        int k = kc * 4 + 2 * hh;              // lanes 0-15: K,K+1 ; 16-31: K+2,K+3
        v2f bS, bN;
        bS.x = Wself[k * H + col];    bS.y = Wself[(k + 1) * H + col];
        bN.x = Wneigh[k * H + col];   bN.y = Wneigh[(k + 1) * H + col];
        #pragma unroll
        for (int s = 0; s < NSUB; ++s) {
            int r = s * 16 + lr;
            v2f a;
            a.x = Ash[r][k];  a.y = Ash[r][k + 1];
            cs[s] = __builtin_amdgcn_wmma_f32_16x16x4_f32(false, a, false, bS,
                                                          (short)0, cs[s], false, false);
            a.x = Nsh[r][k];  a.y = Nsh[r][k + 1];
            cn[s] = __builtin_amdgcn_wmma_f32_16x16x4_f32(false, a, false, bN,
                                                          (short)0, cn[s], false, false);
        }
    }
    #pragma unroll
    for (int s = 0; s < NSUB; ++s)
        #pragma unroll
        for (int v = 0; v < 8; ++v)
            Osh[s * 16 + v + 8 * hh][col] = cs[s][v] + cn[s][v];
    __syncthreads();

    // LayerNorm: 16 threads (one half-warp group) per row, 8 cols each, 5 passes.
    for (int p = 0; p < NSUB; ++p) {
        int rr = p * 16 + (t >> 4);
        int c0 = (t & 15) * 8;
        float xv[8];
        float sum = 0.0f;
        #pragma unroll
        for (int j = 0; j < 8; ++j) {
            int c = c0 + j;
            xv[j] = Osh[rr][c] + bsh[c];
            sum += xv[j];
        }
        for (int m = 1; m < 16; m <<= 1) sum += __shfl_xor(sum, m, 32);
        float mu = sum * (1.0f / H);
        float vs = 0.0f;
        #pragma unroll
        for (int j = 0; j < 8; ++j) { float d = xv[j] - mu; vs += d * d; }
        for (int m = 1; m < 16; m <<= 1) vs += __shfl_xor(vs, m, 32);
        float rstd = rsqrtf(vs * (1.0f / H) + LN_EPS);

        size_t base = (size_t)(node0 + rr) * H;
        #pragma unroll
        for (int j = 0; j < 8; ++j) {
            int c = c0 + j;
            float y = (xv[j] - mu) * rstd * gamma[c] + beta[c];
            if (relu_residual) y = fmaxf(y, 0.0f) + Ash[rr][c];  // residual from LDS
            out[base + c] = y;
        }
    }
}

extern "C" void kernel_launch(void* const* d_in, const int* in_sizes, int n_in,
                              void* d_out, int out_size, void* d_ws, size_t ws_size,
                              hipStream_t stream) {
    (void)in_sizes; (void)n_in; (void)out_size; (void)ws_size;
    const float* x    = (const float*)d_in[0];
    const int*   ei   = (const int*)  d_in[1];
    const float* w_in = (const float*)d_in[2];
    const float* b_in = (const float*)d_in[3];
    const float* ws0  = (const float*)d_in[4];
    const float* bs0  = (const float*)d_in[5];
    const float* wn0  = (const float*)d_in[6];
    const float* bn0  = (const float*)d_in[7];
    const float* g0   = (const float*)d_in[8];
    const float* be0  = (const float*)d_in[9];
    const float* ws1  = (const float*)d_in[10];
    const float* bs1  = (const float*)d_in[11];
    const float* wn1  = (const float*)d_in[12];
    const float* bn1  = (const float*)d_in[13];
    const float* g1   = (const float*)d_in[14];
    const float* be1  = (const float*)d_in[15];
    const int* src = ei;            // edge_index[0]
    const int* tgt = ei + N_EDGES;  // edge_index[1]

    // workspace: h0 | h1 | agg | cnt  (~77 MB)
    float* h0  = (float*)d_ws;
    float* h1  = h0  + (size_t)N_NODES * H;
    float* agg = h1  + (size_t)N_NODES * H;
    float* cnt = agg + (size_t)N_NODES * H;

    dim3 blk(256);
    const int NH = N_NODES * H;
    const int NB_PROJ  = N_NODES / 16;    // 3125
    const int NB_LAYER = N_NODES / ROWS;  // 625

    // degree counts (shared by both layers)
    zero_f32<<<(N_NODES + 255) / 256, blk, 0, stream>>>(cnt, N_NODES);
    count_edges<<<(N_EDGES + 255) / 256, blk, 0, stream>>>(tgt, cnt);

    // h0 = relu(x @ w_in + b_in)
    proj_relu<<<NB_PROJ, blk, 0, stream>>>(x, w_in, b_in, h0);

    // ---- layer 0: gcn -> LN -> relu -> +residual ----
    zero_f32<<<(NH + 255) / 256, blk, 0, stream>>>(agg, NH);
    edge_aggregate<<<(N_EDGES * 32) / 256, blk, 0, stream>>>(h0, src, tgt, agg);
    gcn_layer<<<NB_LAYER, blk, 0, stream>>>(h0, agg, cnt, ws0, bs0, wn0, bn0,
                                            g0, be0, h1, 1);

    // ---- layer 1: gcn -> LN ----
    zero_f32<<<(NH + 255) / 256, blk, 0, stream>>>(agg, NH);
    edge_aggregate<<<(N_EDGES * 32) / 256, blk, 0, stream>>>(h1, src, tgt, agg);
    gcn_layer<<<NB_LAYER, blk, 0, stream>>>(h1, agg, cnt, ws1, bs1, wn1, bn1,
                                            g1, be1, (float*)d_out, 0);
}